// PotentialLoss_88570815578429
// MI455X (gfx1250) — compile-verified
//
#include <hip/hip_runtime.h>
#include <math.h>
#include <stdint.h>

// Object condensation loss on gfx1250.
// Heavy part = [511 x 200000] pairwise block from an M=512,K=8 fp32 GEMM
// (x_alpha @ x^T) done with chained V_WMMA_F32_16X16X4_F32, fused with the
// elementwise attractive/repulsive terms and the masked reduction.
// Whole working set (~9 MB) is L2-resident on MI455X (192 MB L2), so the
// kernel is elementwise-VALU bound; sqrt uses raw v_sqrt_f32 to minimize
// the per-tile VALU tail around the two WMMAs.

#define DIMS   8
#define NPID   512          // particle ids including 0 (noise)
#define NCLUS  511          // real clusters (pids 1..511)
#define CT     32           // cluster tiles: 512/16
#define GYB    16           // blockIdx.y extent
#define WPB    8            // waves per block (256 threads, wave32)
#define NPART  (CT * GYB * WPB)   // 4096 per-wave partials

typedef __attribute__((ext_vector_type(2))) float v2f;
typedef __attribute__((ext_vector_type(8))) float v8f;

// ---------------- kernel 1: init argmax keys ----------------
__global__ void k_init(unsigned long long* keys) {
    int i = blockIdx.x * blockDim.x + threadIdx.x;
    // key = (q_bits << 32) | ~index. Init: q_bits=0, index=0 -> empty cluster
    // decodes to alpha=0, matching jnp.argmax over all -inf.
    if (i < NPID) keys[i] = 0xFFFFFFFFull;
}

// ---------------- kernel 2: per-hit prep + packed argmax ----------------
__global__ void k_prep(const float* __restrict__ beta,
                       const float* __restrict__ x,
                       const int*   __restrict__ pid,
                       float* __restrict__ q,
                       float* __restrict__ xsq,
                       unsigned long long* __restrict__ keys,
                       int n) {
    for (int i = blockIdx.x * blockDim.x + threadIdx.x; i < n;
         i += gridDim.x * blockDim.x) {
        float a  = atanhf(beta[i]);
        float qi = a * a + 0.01f;
        q[i] = qi;
        const float4* xr = (const float4*)(x + (size_t)i * DIMS);
        float4 x0 = xr[0], x1 = xr[1];
        xsq[i] = x0.x*x0.x + x0.y*x0.y + x0.z*x0.z + x0.w*x0.w +
                 x1.x*x1.x + x1.y*x1.y + x1.z*x1.z + x1.w*x1.w;
        int p = pid[i];
        if (p != 0) {
            // q >= 0.01 > 0, so float bit pattern is order-preserving.
            unsigned long long key =
                ((unsigned long long)__float_as_uint(qi) << 32) |
                (unsigned long long)(~(unsigned int)i);
            atomicMax(&keys[p], key);
        }
    }
}

// ---------------- kernel 3: gather condensation points ----------------
__global__ void k_gather(const unsigned long long* __restrict__ keys,
                         const float* __restrict__ x,
                         const float* __restrict__ q,
                         const float* __restrict__ xsq,
                         float* __restrict__ xa,
                         float* __restrict__ qa,
                         float* __restrict__ xasq) {
    int p = blockIdx.x * blockDim.x + threadIdx.x;   // cluster row 0..511
    if (p >= NPID) return;
    if (p < NCLUS) {
        unsigned long long key = keys[p + 1];
        int alpha = (int)(~(unsigned int)(key & 0xFFFFFFFFull));
        #pragma unroll
        for (int d = 0; d < DIMS; ++d)
            xa[p * DIMS + d] = x[(size_t)alpha * DIMS + d];
        qa[p]   = q[alpha];
        xasq[p] = xsq[alpha];
    } else {                // padding row: q_alpha = 0 -> zero contribution
        #pragma unroll
        for (int d = 0; d < DIMS; ++d) xa[p * DIMS + d] = 0.0f;
        qa[p]   = 0.0f;
        xasq[p] = 0.0f;
    }
}

// ---------------- kernel 4: WMMA pairwise + fused loss ----------------
__global__ void __launch_bounds__(256)
k_main(const float* __restrict__ x,
       const int*   __restrict__ pid,
       const float* __restrict__ q,
       const float* __restrict__ xsq,
       const float* __restrict__ xa,
       const float* __restrict__ qa,
       const float* __restrict__ xasq,
       float* __restrict__ partials,
       int n, int ntiles) {
    const int lane = threadIdx.x & 31;
    const int wave = threadIdx.x >> 5;
    const int m_base = blockIdx.x * 16;       // this wave's 16 cluster rows
    const int half = lane >> 4;               // 0: lanes 0-15, 1: lanes 16-31
    const int k0 = half * 2;                  // K offset per ISA A/B layout

    // A operand (16x4 f32, 2 VGPRs): lane holds row (lane&15), K = k0,k0+1.
    const int r = m_base + (lane & 15);
    const float* ar = xa + (size_t)r * DIMS;
    v2f a0 = *(const v2f*)(ar + k0);          // K = k0, k0+1
    v2f a1 = *(const v2f*)(ar + k0 + 4);      // K = k0+4, k0+5

    // C/D layout: acc element v <-> cluster row m_base + v + 8*half.
    const int moff = m_base + half * 8;
    float qa_r[8], xasq_r[8];
    int pid_r[8];
    #pragma unroll
    for (int v = 0; v < 8; ++v) {
        qa_r[v]   = qa[moff + v];
        xasq_r[v] = xasq[moff + v];
        pid_r[v]  = moff + v + 1;             // >= 512 for padding, never matches
    }

    float acc = 0.0f;
    const int hw = blockIdx.y * WPB + wave;   // uniform within wave
    for (int t = hw; t < ntiles; t += GYB * WPB) {
        int nidx = t * 16 + (lane & 15);
        int nc = (nidx < n) ? nidx : (n - 1); // clamp (branchless, EXEC stays full)
        const float* xr = x + (size_t)nc * DIMS;
        v2f b0 = *(const v2f*)(xr + k0);      // B 4x16: lane holds col (lane&15)
        v2f b1 = *(const v2f*)(xr + k0 + 4);
        float qn   = (nidx < n) ? q[nc] : 0.0f;   // kills tail contributions
        float xsqn = xsq[nc];
        int   pn   = pid[nc];

        v8f s = {};
        s = __builtin_amdgcn_wmma_f32_16x16x4_f32(false, a0, false, b0,
                                                  (short)0, s, false, false);
        s = __builtin_amdgcn_wmma_f32_16x16x4_f32(false, a1, false, b1,
                                                  (short)0, s, false, false);
        #pragma unroll
        for (int v = 0; v < 8; ++v) {
            float d2 = fmaxf(xsqn + xasq_r[v] - 2.0f * s[v], 0.0f);
            // raw v_sqrt_f32 (~1 ulp) instead of IEEE-correction sequence
            float dist = __builtin_amdgcn_sqrtf(fmaxf(d2, 1e-12f));
            float hinge = fmaxf(1.0f - dist, 0.0f);
            // q_n*(M*va + 10*(1-M)*vr) == (q_n*q_a) * (M ? d2 : 10*hinge)
            float w = qn * qa_r[v];
            float term = (pn == pid_r[v]) ? d2 : 10.0f * hinge;
            acc = fmaf(w, term, acc);
        }
    }
    acc *= 1.0f / (float)n;                   // mean over N (sum over p folded in)

    // wave32 reduction
    #pragma unroll
    for (int o = 16; o > 0; o >>= 1) acc += __shfl_xor(acc, o, 32);
    if (lane == 0)
        partials[(blockIdx.x * GYB + blockIdx.y) * WPB + wave] = acc;
}

// ---------------- kernel 5: deterministic final reduction ----------------
__global__ void k_reduce(const float* __restrict__ partials, float* __restrict__ out) {
    __shared__ float sm[256];
    float s = 0.0f;
    for (int i = threadIdx.x; i < NPART; i += 256) s += partials[i];  // fixed order
    sm[threadIdx.x] = s;
    __syncthreads();
    for (int st = 128; st > 0; st >>= 1) {
        if (threadIdx.x < st) sm[threadIdx.x] += sm[threadIdx.x + st];
        __syncthreads();
    }
    if (threadIdx.x == 0) out[0] = sm[0];
}

extern "C" void kernel_launch(void* const* d_in, const int* in_sizes, int n_in,
                              void* d_out, int out_size, void* d_ws, size_t ws_size,
                              hipStream_t stream) {
    (void)n_in; (void)out_size; (void)ws_size;
    const int n = in_sizes[0];                       // N hits
    const float* beta = (const float*)d_in[1];
    const float* x    = (const float*)d_in[2];
    const int*   pid  = (const int*)d_in[4];

    // workspace carve-out
    char* ws = (char*)d_ws;
    float* q   = (float*)ws;  ws += sizeof(float) * (size_t)n;
    float* xsq = (float*)ws;  ws += sizeof(float) * (size_t)n;
    uintptr_t u = ((uintptr_t)ws + 7u) & ~(uintptr_t)7u;
    ws = (char*)u;
    unsigned long long* keys = (unsigned long long*)ws; ws += 8 * NPID;
    float* xa       = (float*)ws; ws += sizeof(float) * NPID * DIMS;
    float* qa       = (float*)ws; ws += sizeof(float) * NPID;
    float* xasq     = (float*)ws; ws += sizeof(float) * NPID;
    float* partials = (float*)ws; ws += sizeof(float) * NPART;

    const int ntiles = (n + 15) / 16;

    k_init<<<2, 256, 0, stream>>>(keys);
    k_prep<<<(n + 255) / 256, 256, 0, stream>>>(beta, x, pid, q, xsq, keys, n);
    k_gather<<<2, 256, 0, stream>>>(keys, x, q, xsq, xa, qa, xasq);
    k_main<<<dim3(CT, GYB), 256, 0, stream>>>(x, pid, q, xsq, xa, qa, xasq,
                                              partials, n, ntiles);
    k_reduce<<<1, 256, 0, stream>>>(partials, (float*)d_out);
}